// ConstantCurrentLIFEncoder_10436770530031
// MI455X (gfx1250) — compile-verified
//
#include <hip/hip_runtime.h>

// ---------------------------------------------------------------------------
// constant_current_lif_encode for MI455X (gfx1250)
//
// Output = 100 MB of fp32 spikes -> pure store-bandwidth kernel
// (floor ~4.5us @ 23.3 TB/s). Compute (~3 VALU ops/elem/step) is free.
//
// Strategy:
//  * Split SEQ_LEN=100 into 5 time slices of 20 steps. A block recomputes the
//    membrane state up to its slice start (constant input => deterministic,
//    <=80 discarded FMA steps) -> 1280 independent blocks fill the chip.
//  * Each block (256 thr x 4 neurons) writes its 20x1024 fp32 spike tile to
//    LDS (80 KB), then wave 0 issues ONE Tensor-Data-Mover store
//    (tensor_store_from_lds) that DMAs the strided 2D tile to global memory.
//    This is the CDNA5 bulk data-movement path: one descriptor replaces
//    5120 per-thread global_store_b128s per block.
// ---------------------------------------------------------------------------

#define SEQ_LEN 100
#define TC      20                 // timesteps per block (time slice)
#define TSLICES (SEQ_LEN / TC)     // 5
#define NPT     4                  // neurons per thread
#define THREADS 256
#define NPB     (THREADS * NPT)    // 1024 neurons per block
#define LDS_BYTES (TC * NPB * 4)   // 80 KB tile -> 4 blocks / WGP (320 KB LDS)

typedef unsigned int v4u __attribute__((ext_vector_type(4)));
typedef int          v8i __attribute__((ext_vector_type(8)));
typedef int          v4i __attribute__((ext_vector_type(4)));

__global__ void __launch_bounds__(THREADS)
lif_encode_tdm(const float* __restrict__ in, float* __restrict__ out, int n) {
  // Keep the v-update unfused so arithmetic matches the reference exactly.
  #pragma clang fp contract(off)

  extern __shared__ float smem[];            // [TC][NPB] spike tile

  const int ts  = blockIdx.x % TSLICES;      // time slice index
  const int nb  = blockIdx.x / TSLICES;      // neuron block index
  const int tid = threadIdx.x;
  const int base = nb * NPB + tid * NPT;
  const int t0  = ts * TC;

  // Load 4 constant currents per thread (one coalesced b128 per lane).
  const float4 cur4 = *(const float4*)(in + base);
  float c[NPT] = {cur4.x, cur4.y, cur4.z, cur4.w};
  float v[NPT] = {0.0f, 0.0f, 0.0f, 0.0f};

  // Fast-forward membrane state to t0 (no stores; VALU is free here).
  for (int t = 0; t < t0; ++t) {
    #pragma unroll
    for (int j = 0; j < NPT; ++j) {
      float vn = v[j] + 0.1f * (c[j] - v[j]);   // v += DT*TAU_MEM_INV*(I - v)
      v[j] = (vn > 1.0f) ? 0.0f : vn;           // reset-to-0 on spike
    }
  }

  // Produce this block's TC x NPB spike tile into LDS.
  #pragma unroll
  for (int tt = 0; tt < TC; ++tt) {
    float z[NPT];
    #pragma unroll
    for (int j = 0; j < NPT; ++j) {
      float vn = v[j] + 0.1f * (c[j] - v[j]);
      z[j] = (vn > 1.0f) ? 1.0f : 0.0f;         // heaviside(v - v_th)
      v[j] = (vn > 1.0f) ? 0.0f : vn;
    }
    *(float4*)&smem[tt * NPB + tid * NPT] = make_float4(z[0], z[1], z[2], z[3]);
  }

  __syncthreads();   // all ds_store done (compiler emits s_wait_dscnt + barrier)

  // Wave 0 issues one TDM store for the whole tile:
  //   2D tile: tile_dim0 = 1024 fp32 (contiguous), tile_dim1 = 20 rows,
  //   row stride = n elements, dest = out + (t0*n + nb*NPB).
  if (tid < 32) {
    unsigned long long ga =
        (unsigned long long)(uintptr_t)(out + (size_t)t0 * (size_t)n
                                            + (size_t)nb * NPB);
    // ---- D# group 0 (128b): count / lds_addr / global_addr / type ----
    v4u g0;
    g0.x = 1u;                                   // count=1 valid descriptor
    g0.y = 0u;                                   // lds_addr: dynamic-LDS base
    g0.z = (unsigned)ga;                         // global_addr[31:0]
    g0.w = ((unsigned)(ga >> 32) & 0x1FFFFFFu)   // global_addr[56:32]
           | 0x80000000u;                        // type=2 ("image")

    // ---- D# group 1 (256b): sizes / strides ----
    v8i g1;
    g1[0] = 0x00020000;                          // wg_mask=0, data_size=2 (4B)
    g1[1] = (int)((unsigned)NPB << 16);          // tensor_dim0[15:0]=1024
    g1[2] = (int)((unsigned)TC  << 16);          // dim0 hi=0 | tensor_dim1=20
    g1[3] = (int)((unsigned)NPB << 16);          // dim1 hi=0 | tile_dim0=1024
    g1[4] = TC;                                  // tile_dim1=20, tile_dim2=0
    g1[5] = n;                                   // tensor_dim0_stride[31:0]
    g1[6] = 0;                                   // stride0 hi | stride1 lo
    g1[7] = 0;                                   // stride1 hi

    v4i g2 = {0, 0, 0, 0};                       // 2D tensor: groups 2/3 unused
    v4i g3 = {0, 0, 0, 0};
    v8i g4 = {0, 0, 0, 0, 0, 0, 0, 0};           // extended group (unused, 2D)

    // clang-23 / therock-10.0 signature: 6 args (…, int32x8, i32 cpol)
    __builtin_amdgcn_tensor_store_from_lds(g0, g1, g2, g3, g4, /*cpol=*/0);
    __builtin_amdgcn_s_wait_tensorcnt(0);
  }
}

extern "C" void kernel_launch(void* const* d_in, const int* in_sizes, int n_in,
                              void* d_out, int out_size, void* d_ws, size_t ws_size,
                              hipStream_t stream) {
  (void)n_in; (void)out_size; (void)d_ws; (void)ws_size;
  const float* in  = (const float*)d_in[0];
  float*       out = (float*)d_out;
  const int n = in_sizes[0];                     // 262144 neurons
  const int nblocks = (n / NPB) * TSLICES;       // 256 * 5 = 1280 blocks
  lif_encode_tdm<<<nblocks, THREADS, LDS_BYTES, stream>>>(in, out, n);
}